// ToeplitzMatrix_64716567216716
// MI455X (gfx1250) — compile-verified
//
#include <hip/hip_runtime.h>

// ---------------------------------------------------------------------------
// out[b,s,o] = sum_i x[b,s,i] * T[o,i] + bias[o],  T[o,i] = g[o-i]
//   g[d] = d>=0 ? first_col[d] : first_row[-d]
// GEMM: M=32768, K=1024, N=1024, fp32 -> memory-bound on MI455X.
// fp32 WMMA (v_wmma_f32_16x16x4_f32); Toeplitz window h[] kept in LDS
// (ascending-k order so B pairs load contiguously, no swap movs);
// x streamed via gfx1250 GLOBAL_LOAD_ASYNC_TO_LDS_B128 (ASYNCcnt) when the
// builtin exists, else register-staged b128 loads + ds stores.
// ---------------------------------------------------------------------------

#define K_DIM 1024
#define N_DIM 1024
#define BM 128         // block tile M
#define BN 128         // block tile N
#define KT 32          // K tile
#define KTP 36         // padded LDS row stride (dwords): 16B-aligned rows, bank-spread
#define GLEN (BN + K_DIM)      // 1152: j = k - n_local + (BN-1), k<=1023(+1), n_local<128

typedef __attribute__((ext_vector_type(2))) float v2f;
typedef __attribute__((ext_vector_type(8))) float v8f;
typedef int v4i_vs __attribute__((vector_size(16)));   // matches builtin param type

#if __has_builtin(__builtin_amdgcn_global_load_async_to_lds_b128)
#define USE_ASYNC_LDS 1
#else
#define USE_ASYNC_LDS 0
#endif

__global__ __launch_bounds__(256)
void toeplitz_wmma_kernel(const float* __restrict__ X,
                          const float* __restrict__ first_row,
                          const float* __restrict__ first_col,
                          const float* __restrict__ bias,
                          float* __restrict__ Out)
{
    __shared__ float hbuf[GLEN];              // Toeplitz window, ascending in k
    __shared__ float xs[2][BM * KTP];         // 2 x 18 KB x-tile double buffer

    const int tid  = threadIdx.x;
    const int lane = tid & 31;
    const int wid  = tid >> 5;                // 8 waves
    const int wm   = wid >> 2;                // 0..1  (wave tile row)
    const int wn   = wid & 3;                 // 0..3  (wave tile col)
    const int l15  = lane & 15;
    const int lh   = lane >> 4;               // 0/1: half-wave selects K pair / M half

    const int n0 = blockIdx.x * BN;
    const int m0 = blockIdx.y * BM;

    // ---- Toeplitz window: hbuf[j] = W[k,n] with j = k - n_local + (BN-1) ----
    // W[k,n] = T[n,k] = g[n-k]; d = n - k = n0 + (BN-1) - j
    for (int j = tid; j < GLEN; j += 256) {
        int d = n0 + (BN - 1) - j;
        hbuf[j] = (d >= 0) ? first_col[d] : first_row[-d];
    }

#if USE_ASYNC_LDS
    // ---- gfx1250 async copy: global -> LDS, no VGPR staging, ASYNCcnt ----
    auto load_tile = [&](int k0, int buf) {
        #pragma unroll
        for (int r = 0; r < 4; ++r) {
            int idx  = tid + r * 256;         // 0..1023
            int row  = idx >> 3;              // 0..127
            int quad = idx & 7;               // 16B chunks
            __builtin_amdgcn_global_load_async_to_lds_b128(
                (__attribute__((address_space(1))) v4i_vs*)
                    &X[(size_t)(m0 + row) * K_DIM + k0 + quad * 4],
                (__attribute__((address_space(3))) v4i_vs*)
                    &xs[buf][row * KTP + quad * 4],
                0, 0);
        }
    };
    auto wait_tile = [&]() {
#if __has_builtin(__builtin_amdgcn_s_wait_asynccnt)
        __builtin_amdgcn_s_wait_asynccnt(0);
#else
        asm volatile("s_wait_asynccnt 0" ::: "memory");
#endif
    };
    load_tile(0, 0);
    wait_tile();
    __syncthreads();
#else
    // ---- fallback: register-staged b128 loads + LDS stores ----
    float4 pf[4];
    auto fetch_tile = [&](int k0) {
        #pragma unroll
        for (int r = 0; r < 4; ++r) {
            int idx  = tid + r * 256;
            int row  = idx >> 3;
            int quad = idx & 7;
            pf[r] = *(const float4*)&X[(size_t)(m0 + row) * K_DIM + k0 + quad * 4];
        }
    };
    auto commit_tile = [&](int buf) {
        #pragma unroll
        for (int r = 0; r < 4; ++r) {
            int idx  = tid + r * 256;
            int row  = idx >> 3;
            int quad = idx & 7;
            *(float4*)&xs[buf][row * KTP + quad * 4] = pf[r];
        }
    };
    fetch_tile(0);
    commit_tile(0);
    __syncthreads();
#endif

    // ---- accumulators: wave tile 64x32 = 4x2 WMMA 16x16 tiles ----
    v8f acc[4][2];
    #pragma unroll
    for (int mi = 0; mi < 4; ++mi)
        #pragma unroll
        for (int ni = 0; ni < 2; ++ni)
            acc[mi][ni] = (v8f){0.f, 0.f, 0.f, 0.f, 0.f, 0.f, 0.f, 0.f};

    const int nt = K_DIM / KT;                // 32 K-tiles
    for (int t = 0; t < nt; ++t) {
#if USE_ASYNC_LDS
        if (t + 1 < nt) load_tile((t + 1) * KT, (t + 1) & 1);
#else
        if (t + 1 < nt) fetch_tile((t + 1) * KT);
#endif
        const float* __restrict__ xsb = xs[t & 1];

        #pragma unroll
        for (int kk = 0; kk < KT; kk += 4) {
            // A fragments (16x4 f32): lane l15 = row, vgpr v holds k = kk + 2*lh + v
            v2f a[4];
            #pragma unroll
            for (int mi = 0; mi < 4; ++mi) {
                int row = wm * 64 + mi * 16 + l15;
                a[mi] = *(const v2f*)&xsb[row * KTP + kk + 2 * lh];  // 8B aligned
            }
            // B fragments (4x16 f32): ascending contiguous pair in hbuf
            v2f b[2];
            #pragma unroll
            for (int ni = 0; ni < 2; ++ni) {
                int ncol = wn * 32 + ni * 16 + l15;         // n within block tile
                int kg   = t * KT + kk + 2 * lh;            // global k for vgpr0
                int ji   = kg + (BN - 1) - ncol;
                b[ni] = (v2f){hbuf[ji], hbuf[ji + 1]};      // k, k+1: contiguous
            }
            // 8 outer-product WMMAs
            #pragma unroll
            for (int mi = 0; mi < 4; ++mi)
                #pragma unroll
                for (int ni = 0; ni < 2; ++ni)
                    acc[mi][ni] = __builtin_amdgcn_wmma_f32_16x16x4_f32(
                        false, a[mi], false, b[ni], (short)0, acc[mi][ni],
                        false, false);
        }

#if USE_ASYNC_LDS
        wait_tile();                          // our async writes landed in LDS
        __syncthreads();                      // visible to all waves
#else
        if (t + 1 < nt) commit_tile((t + 1) & 1);
        __syncthreads();
#endif
    }

    // ---- epilogue: C/D layout: vgpr r -> row (lh*8 + r), lane l15 -> col ----
    #pragma unroll
    for (int mi = 0; mi < 4; ++mi) {
        int rowbase = m0 + wm * 64 + mi * 16 + lh * 8;
        #pragma unroll
        for (int ni = 0; ni < 2; ++ni) {
            int ncol = n0 + wn * 32 + ni * 16 + l15;
            float bv = bias[ncol];
            #pragma unroll
            for (int r = 0; r < 8; ++r) {
                Out[(size_t)(rowbase + r) * N_DIM + ncol] = acc[mi][ni][r] + bv;
            }
        }
    }
}

extern "C" void kernel_launch(void* const* d_in, const int* in_sizes, int n_in,
                              void* d_out, int out_size, void* d_ws, size_t ws_size,
                              hipStream_t stream) {
    const float* x    = (const float*)d_in[0];
    const float* frow = (const float*)d_in[1];
    const float* fcol = (const float*)d_in[2];
    const float* bias = (const float*)d_in[3];
    float* out = (float*)d_out;

    const int M = in_sizes[0] / K_DIM;        // 4*8192 = 32768 rows
    dim3 grid(N_DIM / BN, M / BM);            // (8, 256)
    toeplitz_wmma_kernel<<<grid, 256, 0, stream>>>(x, frow, fcol, bias, out);
}